// CPPN_72919954751874
// MI455X (gfx1250) — compile-verified
//
#include <hip/hip_runtime.h>

typedef __attribute__((ext_vector_type(2))) float v2f;
typedef __attribute__((ext_vector_type(8))) float v8f;

#define N_LAYERS 12
#define DH 22          // hidden width
#define KPAD 24        // K padded to 6 chunks of 4
#define NCH 6          // number of K-chunks
#define LSTRIDE 768    // packed floats per hidden-layer weight block
#define HSTRIDE 24     // sH row stride in floats

__device__ __forceinline__ float activate(float v, int col) {
  // cols 0..14: identity, 15..18: gaussian, 19..20: identity, 21: sin
  float g = __expf(-v * v) * 2.0f - 1.0f;
  float s = __sinf(v);
  float r = (col >= 15 && col < 19) ? g : v;
  r = (col == 21) ? s : r;
  return r;
}

__global__ __launch_bounds__(256) void cppn_wmma_kernel(
    const float* __restrict__ x,
    const float* __restrict__ w_in,
    const float* __restrict__ w_h,
    const float* __restrict__ w_out,
    float* __restrict__ out,
    int N) {
  __shared__ __align__(16) float sPBin[128];           // layer-0 B operands
  __shared__ __align__(16) float sPBh[11 * LSTRIDE];   // hidden-layer B operands
  __shared__ __align__(16) float sPBout[384];          // output-layer B operands
  __shared__ __align__(16) float sH[8 * 16 * HSTRIDE]; // per-wave H staging tile

  const int tid = threadIdx.x;

  // ---- pack weights into exact WMMA B-operand lane order (zero-padded) ----
  // reader index: ((nt*NCH + k)*32 + lane)*2 + j  ->  W[4k + 2*(lane>>4) + j][nt*16 + (lane&15)]
  for (int q = tid; q < 128; q += 256) {
    int j = q & 1, lane = (q >> 1) & 31, nt = q >> 6;
    int r = 2 * (lane >> 4) + j;
    int c = nt * 16 + (lane & 15);
    sPBin[q] = (c < DH) ? w_in[r * DH + c] : 0.0f;
  }
  for (int q = tid; q < 11 * LSTRIDE; q += 256) {
    int i = q / LSTRIDE, t = q - i * LSTRIDE;
    int j = t & 1, lane = (t >> 1) & 31, kc = (t >> 6) % NCH, nt = t / (NCH * 64);
    int r = 4 * kc + 2 * (lane >> 4) + j;
    int c = nt * 16 + (lane & 15);
    sPBh[q] = (r < DH && c < DH) ? w_h[(i * DH + r) * DH + c] : 0.0f;
  }
  for (int q = tid; q < 384; q += 256) {
    int j = q & 1, lane = (q >> 1) & 31, kc = q >> 6;
    int r = 4 * kc + 2 * (lane >> 4) + j;
    int c = lane & 15;
    sPBout[q] = (r < DH && c < 3) ? w_out[r * 3 + c] : 0.0f;
  }

  // ---- x passthrough feature (feature slot #1) ----
  {
    size_t xbase = (size_t)3 * N;
    int base = blockIdx.x * 512;
    for (int q = tid; q < 512; q += 256) {
      int idx = base + q;
      out[xbase + idx] = x[idx];
    }
  }

  __syncthreads();

  const int lane = tid & 31;
  const int wave = tid >> 5;
  const int lhalf = lane >> 4;   // 0: low half-wave, 1: high half-wave
  const int lm = lane & 15;      // A-row (M) for A operand; column (N) for B/D
  const int p0 = blockIdx.x * 128 + wave * 16;
  float* hw = &sH[wave * 16 * HSTRIDE];

  const size_t hbase = (size_t)7 * N;    // start of h feature slots
  const size_t o2 = (size_t)271 * N;     // second copy of out

  // ---- layer 0: [16x4] x [4x22], A loaded straight from x ----
  {
    v2f a0 = *(const v2f*)(x + (size_t)(p0 + lm) * 4 + 2 * lhalf);
#pragma unroll
    for (int nt = 0; nt < 2; ++nt) {
      v8f acc = {};
      v2f b = *(const v2f*)(&sPBin[(nt * 32 + lane) * 2]);
      acc = __builtin_amdgcn_wmma_f32_16x16x4_f32(false, a0, false, b,
                                                  (short)0, acc, false, false);
      int col = nt * 16 + lm;
#pragma unroll
      for (int r8 = 0; r8 < 8; ++r8) {
        float v = activate(acc[r8], col);
        int row = r8 + 8 * lhalf;
        if (col < KPAD) hw[row * HSTRIDE + col] = v;
        if (col < DH) out[hbase + (size_t)(p0 + row) * DH + col] = v;
      }
    }
  }

  // ---- layers 1..11: [16x22] x [22x22] via 2 N-tiles x 6 chained K-steps ----
  for (int layer = 1; layer < N_LAYERS; ++layer) {
    v2f a[NCH];
#pragma unroll
    for (int k = 0; k < NCH; ++k)
      a[k] = *(const v2f*)(&hw[lm * HSTRIDE + 4 * k + 2 * lhalf]);
    const float* pb = &sPBh[(layer - 1) * LSTRIDE];
    size_t hb = hbase + (size_t)layer * DH * N;
#pragma unroll
    for (int nt = 0; nt < 2; ++nt) {
      v8f acc = {};
#pragma unroll
      for (int k = 0; k < NCH; ++k) {
        v2f b = *(const v2f*)(&pb[((nt * NCH + k) * 32 + lane) * 2]);
        acc = __builtin_amdgcn_wmma_f32_16x16x4_f32(false, a[k], false, b,
                                                    (short)0, acc, false, false);
      }
      int col = nt * 16 + lm;
#pragma unroll
      for (int r8 = 0; r8 < 8; ++r8) {
        float v = activate(acc[r8], col);
        int row = r8 + 8 * lhalf;
        if (col < KPAD) hw[row * HSTRIDE + col] = v;
        if (col < DH) out[hb + (size_t)(p0 + row) * DH + col] = v;
      }
    }
  }

  // ---- output layer: [16x22] x [22x3] (single N-tile) ----
  {
    v2f a[NCH];
#pragma unroll
    for (int k = 0; k < NCH; ++k)
      a[k] = *(const v2f*)(&hw[lm * HSTRIDE + 4 * k + 2 * lhalf]);
    v8f acc = {};
#pragma unroll
    for (int k = 0; k < NCH; ++k) {
      v2f b = *(const v2f*)(&sPBout[(k * 32 + lane) * 2]);
      acc = __builtin_amdgcn_wmma_f32_16x16x4_f32(false, a[k], false, b,
                                                  (short)0, acc, false, false);
    }
    if (lm < 3) {
#pragma unroll
      for (int r8 = 0; r8 < 8; ++r8) {
        int row = r8 + 8 * lhalf;
        float v = acc[r8];
        out[(size_t)(p0 + row) * 3 + lm] = v;
        out[o2 + (size_t)(p0 + row) * 3 + lm] = v;
      }
    }
  }
}

extern "C" void kernel_launch(void* const* d_in, const int* in_sizes, int n_in,
                              void* d_out, int out_size, void* d_ws, size_t ws_size,
                              hipStream_t stream) {
  const float* x = (const float*)d_in[0];      // [N,4]
  const float* w_in = (const float*)d_in[1];   // [4,22]
  const float* w_h = (const float*)d_in[2];    // [11,22,22]
  const float* w_out = (const float*)d_in[3];  // [22,3]
  float* out = (float*)d_out;                  // 274*N floats
  int N = in_sizes[0] / 4;                     // 524288
  int blocks = N / 128;                        // 4096 blocks, 8 waves each
  cppn_wmma_kernel<<<blocks, 256, 0, stream>>>(x, w_in, w_h, w_out, out, N);
}